// CCHLoss_85667417686468
// MI455X (gfx1250) — compile-verified
//
#include <hip/hip_runtime.h>
#include <hip/hip_bf16.h>

// Chamfer (single-directional, squared-L2) via V_WMMA_F32_16X16X4_F32.
// d2(i,j) = |x_i|^2 + |y_j|^2 - 2 x_i.y_j
// A row i = (-2xi0, -2xi1, -2xi2, 1);  B col j = (yj0, yj1, yj2, |yj|^2)
// => WMMA tile D = d2 - |x_i|^2 (row-constant shift, argmin-invariant).
//
// Round-3 refinement: keep TWO ds_load_2addr_b64 in flight per inner
// iteration (4 B-tiles). The first WMMA group waits only for dscnt<=1;
// the second load's LDS latency is hidden behind pair-0's 8 WMMAs and
// 32 v_min3s. (A register-rotating prefetch was re-rolled by the
// scheduler in round 2; in-iteration dual loads can't be.)

typedef __attribute__((ext_vector_type(2))) float v2f;
typedef __attribute__((ext_vector_type(8))) float v8f;

constexpr int BATCH = 4;
constexpr int NPTS  = 8192;              // points per cloud
constexpr int CHUNK = 2048;              // y points staged in LDS per pass (32 KB)
constexpr int TILES = CHUNK / 16;        // 128 B-tiles per chunk
constexpr int XT    = 4;                 // 16-row x tiles per wave
constexpr int WAVES = 8;                 // waves per block
constexpr int BLK   = WAVES * 32;        // 256 threads
constexpr int XPTS_PER_BLOCK   = XT * 16 * WAVES;          // 512
constexpr int BLOCKS_PER_BATCH = NPTS / XPTS_PER_BLOCK;    // 16
constexpr int GRID  = BATCH * BLOCKS_PER_BATCH;            // 64
constexpr int NPART = GRID * WAVES;                        // 512 partials (2 KB ws)

// One tile-pair: 8 WMMAs issued before any consumer, then 8 v_min3 per x-tile.
__device__ __forceinline__ void tile_pair_step(const v2f (&a)[XT],
                                               float2 e0, float2 e1,
                                               v8f (&acc)[XT]) {
  v2f b0; b0[0] = e0.x; b0[1] = e0.y;
  v2f b1; b1[0] = e1.x; b1[1] = e1.y;
  v8f d0[XT], d1[XT];
  #pragma unroll
  for (int t = 0; t < XT; ++t) {
    v8f z = {};
    d0[t] = __builtin_amdgcn_wmma_f32_16x16x4_f32(
        false, a[t], false, b0, (short)0, z, false, false);
    d1[t] = __builtin_amdgcn_wmma_f32_16x16x4_f32(
        false, a[t], false, b1, (short)0, z, false, false);
  }
  #pragma unroll
  for (int t = 0; t < XT; ++t)
    #pragma unroll
    for (int r = 0; r < 8; ++r)
      acc[t][r] = fminf(fminf(acc[t][r], d0[t][r]), d1[t][r]);
}

__global__ __launch_bounds__(BLK) void chamfer_min_kernel(
    const float* __restrict__ y,        // v      [B, M, 3] targets
    const float* __restrict__ x,        // v_pred [B, N, 3] queries
    float* __restrict__ partial)        // [NPART] per-wave sums of min d2
{
  __shared__ float4 ytab[CHUNK];        // {y0, y1, y2, |y|^2} per staged point

  const int tid  = threadIdx.x;
  const int lane = tid & 31;
  const int wave = tid >> 5;
  const int l16  = lane & 15;
  const int hi   = lane >> 4;           // 0: lanes 0-15 (K=0,1), 1: lanes 16-31 (K=2,3)

  const int batch  = blockIdx.x / BLOCKS_PER_BATCH;
  const int xblock = blockIdx.x % BLOCKS_PER_BATCH;

  const float* yb = y + (size_t)batch * NPTS * 3;
  const float* xb = x + (size_t)batch * NPTS * 3;

  // ---- Persistent A operands: 16x4 f32 A layout (lanes 0-15: K0/K1, 16-31: K2/K3) ----
  v2f   a[XT];
  float x2s[XT];                        // |x|^2 of point (tile row l16), held by both halves
  const int xbase = xblock * XPTS_PER_BLOCK + wave * (XT * 16);
  #pragma unroll
  for (int t = 0; t < XT; ++t) {
    const float* p = xb + (size_t)(xbase + t * 16 + l16) * 3;
    float c0 = p[0], c1 = p[1], c2 = p[2];
    x2s[t] = c0 * c0 + c1 * c1 + c2 * c2;
    a[t][0] = hi ? (-2.0f * c2) : (-2.0f * c0);
    a[t][1] = hi ? 1.0f         : (-2.0f * c1);
  }

  // ---- Min accumulators (C/D layout: VGPR r -> rows r / r+8, lane -> column) ----
  v8f acc[XT];
  #pragma unroll
  for (int t = 0; t < XT; ++t)
    #pragma unroll
    for (int r = 0; r < 8; ++r)
      acc[t][r] = 3.0e38f;

  const float2* bt   = reinterpret_cast<const float2*>(ytab);
  const int     boff = l16 * 2 + hi;    // lane's element within a 16-point B tile

  for (int c = 0; c < NPTS / CHUNK; ++c) {
    __syncthreads();                    // previous chunk fully consumed
    // Stage CHUNK target points: 8 points/thread, 3x float4 -> 4 points, |y|^2.
    #pragma unroll
    for (int it = 0; it < 2; ++it) {
      const float4* g = reinterpret_cast<const float4*>(
          yb + (size_t)c * CHUNK * 3 + tid * 24 + it * 12);   // 16B aligned
      float4 q0 = g[0], q1 = g[1], q2 = g[2];
      int p = tid * 8 + it * 4;
      ytab[p + 0] = make_float4(q0.x, q0.y, q0.z, q0.x*q0.x + q0.y*q0.y + q0.z*q0.z);
      ytab[p + 1] = make_float4(q0.w, q1.x, q1.y, q0.w*q0.w + q1.x*q1.x + q1.y*q1.y);
      ytab[p + 2] = make_float4(q1.z, q1.w, q2.x, q1.z*q1.z + q1.w*q1.w + q2.x*q2.x);
      ytab[p + 3] = make_float4(q2.y, q2.z, q2.w, q2.y*q2.y + q2.z*q2.z + q2.w*q2.w);
    }
    __syncthreads();

    // Sweep y tiles, 4 per iteration with 2 DS loads in flight.
    // B tile j for this lane: lanes 0-15 -> (y0,y1) of pt l16, 16-31 -> (y2,|y|^2).
    for (int j = 0; j < TILES; j += 4) {
      const float2* base = bt + j * 32 + boff;
      float2 e0 = base[0];              // tiles j, j+1   -> ds_load_2addr_b64 #1
      float2 e1 = base[32];
      float2 e2 = base[64];             // tiles j+2, j+3 -> ds_load_2addr_b64 #2
      float2 e3 = base[96];
      tile_pair_step(a, e0, e1, acc);   // needs only load #1 (dscnt <= 1)
      tile_pair_step(a, e2, e3, acc);   // load #2 latency hidden behind pair 0
    }
  }

  // ---- Column-min over the 16 lanes of each half, add |x|^2, sum rows ----
  float wsum = 0.0f;
  #pragma unroll
  for (int t = 0; t < XT; ++t) {
    #pragma unroll
    for (int r = 0; r < 8; ++r) {
      float v = acc[t][r];
      v = fminf(v, __shfl_xor(v, 1, 32));
      v = fminf(v, __shfl_xor(v, 2, 32));
      v = fminf(v, __shfl_xor(v, 4, 32));
      v = fminf(v, __shfl_xor(v, 8, 32));
      // lanes 0-15 now hold min of row r; lanes 16-31 min of row r+8
      int src = hi ? (r + 8) : r;              // x2s lives in lower-half lane = row
      wsum += v + __shfl(x2s[t], src, 32);
    }
  }
  // lane 0 holds sum over rows 0-7 of each tile, lane 16 over rows 8-15
  float tot = __shfl(wsum, 0, 32) + __shfl(wsum, 16, 32);
  if (lane == 0) partial[blockIdx.x * WAVES + wave] = tot;
}

__global__ __launch_bounds__(NPART) void chamfer_reduce_kernel(
    const float* __restrict__ partial, float* __restrict__ out, float scale)
{
  __shared__ float s[NPART];
  int t = threadIdx.x;
  s[t] = partial[t];
  __syncthreads();
  #pragma unroll
  for (int step = NPART / 2; step > 0; step >>= 1) {
    if (t < step) s[t] += s[t + step];
    __syncthreads();
  }
  if (t == 0) out[0] = s[0] * scale;
}

extern "C" void kernel_launch(void* const* d_in, const int* in_sizes, int n_in,
                              void* d_out, int out_size, void* d_ws, size_t ws_size,
                              hipStream_t stream) {
  (void)in_sizes; (void)n_in; (void)out_size; (void)ws_size;
  const float* v      = (const float*)d_in[0];   // targets  (y)
  const float* v_pred = (const float*)d_in[1];   // queries  (x)
  float* out     = (float*)d_out;
  float* partial = (float*)d_ws;                 // NPART floats = 2 KB

  chamfer_min_kernel<<<GRID, BLK, 0, stream>>>(v, v_pred, partial);
  chamfer_reduce_kernel<<<1, NPART, 0, stream>>>(
      partial, out, 1.0f / (float)(BATCH * NPTS));
}